// AdaptiveDeepSeekSparseAttention_71545565217041
// MI455X (gfx1250) — compile-verified
//
#include <hip/hip_runtime.h>

#define B_SZ 2
#define S_LEN 2048
#define DM 2048
#define NH 16
#define DN 128
#define DR 64
#define DV 128
#define KVR 512
#define HI_N 4
#define DI_N 64
#define NEGV -1000000000.0f

typedef float v2f __attribute__((ext_vector_type(2)));
typedef float v8f __attribute__((ext_vector_type(8)));
typedef unsigned u32x4 __attribute__((ext_vector_type(4)));
typedef int i32x4 __attribute__((ext_vector_type(4)));
typedef int i32x8 __attribute__((ext_vector_type(8)));

#if defined(__has_builtin)
#if __has_builtin(__builtin_amdgcn_tensor_load_to_lds) && \
    __has_builtin(__builtin_amdgcn_s_wait_tensorcnt)
#define USE_TDM 1
#else
#define USE_TDM 0
#endif
#else
#define USE_TDM 0
#endif

__device__ __forceinline__ v8f v8f_zero() {
    v8f z;
#pragma unroll
    for (int i = 0; i < 8; ++i) z[i] = 0.0f;
    return z;
}

// D = A(16x4) x B(4x16) + C, pure fp32 CDNA5 matrix op.
__device__ __forceinline__ v8f wmma_f32(v2f a, v2f b, v8f c) {
    return __builtin_amdgcn_wmma_f32_16x16x4_f32(false, a, false, b, (short)0, c,
                                                 false, false);
}

// ---------------------------------------------------------------------------
// 1) adaptive_k: pooled = mean_s(hs); frac = sigmoid(pooled . w_ctrl + b);
//    k = clip(round(frac*S), 1, S)
// ---------------------------------------------------------------------------
__global__ void ctrl_kernel(const float* __restrict__ hs,
                            const float* __restrict__ w_ctrl,
                            const float* __restrict__ b_ctrl,
                            int* __restrict__ adk) {
    const int b = blockIdx.x, tid = threadIdx.x;
    __shared__ double red[256];
    const float* base = hs + (size_t)b * S_LEN * DM;
    const int total = S_LEN * DM;
    double s = 0.0;
    for (int i = tid; i < total; i += 256)
        s += (double)base[i] * (double)w_ctrl[i & (DM - 1)];
    red[tid] = s;
    __syncthreads();
    for (int st = 128; st > 0; st >>= 1) {
        if (tid < st) red[tid] += red[tid + st];
        __syncthreads();
    }
    if (tid == 0) {
        float x = (float)(red[0] / (double)S_LEN) + b_ctrl[0];
        float frac = 1.0f / (1.0f + expf(-x));
        float kf = rintf(frac * (float)S_LEN);   // round-half-even == jnp.round
        kf = fminf(fmaxf(kf, 1.0f), (float)S_LEN);
        adk[b] = (int)kf;
    }
}

// ---------------------------------------------------------------------------
// 2) fp32 WMMA GEMM: C[M,N] = A[M,K] @ W[K,N] (+bias), row-major.
//    Wave tile: 32M x 64N (2 A-fragments share 4 B-fragments -> 8 WMMAs per
//    k-step). Register double-buffering: k+4 fragments are loaded before the
//    k-step WMMAs issue, so s_wait_loadcnt carries slack instead of 0.
//    Requires M % 32 == 0, N % 64 == 0, K % 4 == 0.
// ---------------------------------------------------------------------------
__global__ void gemm_wmma(const float* __restrict__ A, const float* __restrict__ Bm,
                          const float* __restrict__ bias, float* __restrict__ C,
                          int M, int N, int K) {
    const int lane = threadIdx.x & 31;
    const int wave = threadIdx.x >> 5;
    const int lrow = lane & 15;
    const int lhk  = lane >> 4;
    const int m0 = blockIdx.x * 32;
    const int n0 = blockIdx.y * 256 + wave * 64;
    if (n0 >= N) return;  // wave-uniform: EXEC stays all-ones for WMMA

    const float* arow0 = A + (size_t)(m0 + lrow) * K + 2 * lhk;
    const float* arow1 = arow0 + (size_t)16 * K;
    const float* bbase = Bm + (size_t)(2 * lhk) * N + n0 + lrow;

    v8f acc0[4], acc1[4];
#pragma unroll
    for (int t = 0; t < 4; ++t) { acc0[t] = v8f_zero(); acc1[t] = v8f_zero(); }

    // prologue: fragments for k = 0
    v2f a0 = *(const v2f*)(arow0);
    v2f a1 = *(const v2f*)(arow1);
    v2f bf[4];
#pragma unroll
    for (int t = 0; t < 4; ++t) { bf[t].x = bbase[t * 16]; bf[t].y = bbase[N + t * 16]; }

    for (int k = 4; k < K; k += 4) {
        // issue next-step loads first
        v2f a0n = *(const v2f*)(arow0 + k);
        v2f a1n = *(const v2f*)(arow1 + k);
        const float* bp = bbase + (size_t)k * N;
        v2f bn[4];
#pragma unroll
        for (int t = 0; t < 4; ++t) { bn[t].x = bp[t * 16]; bn[t].y = bp[N + t * 16]; }
        // compute current step
#pragma unroll
        for (int t = 0; t < 4; ++t) acc0[t] = wmma_f32(a0, bf[t], acc0[t]);
#pragma unroll
        for (int t = 0; t < 4; ++t) acc1[t] = wmma_f32(a1, bf[t], acc1[t]);
        a0 = a0n; a1 = a1n;
#pragma unroll
        for (int t = 0; t < 4; ++t) bf[t] = bn[t];
    }
#pragma unroll
    for (int t = 0; t < 4; ++t) acc0[t] = wmma_f32(a0, bf[t], acc0[t]);
#pragma unroll
    for (int t = 0; t < 4; ++t) acc1[t] = wmma_f32(a1, bf[t], acc1[t]);

#pragma unroll
    for (int t = 0; t < 4; ++t) {
#pragma unroll
        for (int j = 0; j < 8; ++j) {
            int col = n0 + t * 16 + lrow;
            float bv = bias ? bias[col] : 0.0f;
            int row0 = m0 + j + 8 * lhk;
            C[(size_t)row0 * N + col] = acc0[t][j] + bv;
            C[(size_t)(row0 + 16) * N + col] = acc1[t][j] + bv;
        }
    }
}

// ---------------------------------------------------------------------------
// 3) index_scores[b,q,k] = sum_h idx_w[h] * relu( qi[b,q,h,:] . ki[b,k,h,:] )
//    One wave per 16x16 score tile; K=64 per head via 16 WMMAs.
// ---------------------------------------------------------------------------
__global__ void index_scores_kernel(const float* __restrict__ qi,
                                    const float* __restrict__ ki,
                                    const float* __restrict__ idx_w,
                                    float* __restrict__ scores) {
    const int lane = threadIdx.x & 31, wave = threadIdx.x >> 5;
    const int lrow = lane & 15, lhk = lane >> 4;
    const int q0 = blockIdx.x * 16;
    const int k0 = (blockIdx.y * 4 + wave) * 16;
    const int b = blockIdx.z;

    const float* qrow = qi + (size_t)(b * S_LEN + q0 + lrow) * (HI_N * DI_N) + 2 * lhk;
    const float* krow = ki + (size_t)(b * S_LEN + k0 + lrow) * (HI_N * DI_N) + 2 * lhk;

    v8f out = v8f_zero();
    for (int h = 0; h < HI_N; ++h) {
        const float wh = idx_w[h];
        const float* qh = qrow + h * DI_N;
        const float* kh = krow + h * DI_N;
        v8f c = v8f_zero();
#pragma unroll
        for (int d = 0; d < DI_N; d += 4) {
            v2f a  = *(const v2f*)(qh + d);
            v2f bf = *(const v2f*)(kh + d);
            c = wmma_f32(a, bf, c);
        }
#pragma unroll
        for (int j = 0; j < 8; ++j) out[j] += fmaxf(c[j], 0.0f) * wh;
    }
#pragma unroll
    for (int j = 0; j < 8; ++j)
        scores[(size_t)(b * S_LEN + q0 + j + 8 * lhk) * S_LEN + k0 + lrow] = out[j];
}

// ---------------------------------------------------------------------------
// 4) Per (b,q) row: stage 2048 scores into LDS (via the Tensor Data Mover
//    where available), bitonic-sort, thr = k-th largest.
// ---------------------------------------------------------------------------
__global__ void topk_thr_kernel(const float* __restrict__ scores,
                                const int* __restrict__ adk,
                                float* __restrict__ thr) {
    const int row = blockIdx.x;   // b*S + q
    const int tid = threadIdx.x;
    __shared__ float buf[S_LEN];

#if USE_TDM
    if (tid == 0) {
        // D# group 0: count=1 | lds_addr | global_addr[56:0] | type=2
        unsigned lds = (unsigned)(uintptr_t)(void*)buf;  // LDS byte offset
        unsigned long long ga =
            (unsigned long long)(uintptr_t)(scores + (size_t)row * S_LEN);
        u32x4 g0;
        g0[0] = 1u;                                       // count=1, no gather
        g0[1] = lds;                                      // lds_addr
        g0[2] = (unsigned)(ga & 0xffffffffu);             // global_addr[31:0]
        g0[3] = (unsigned)((ga >> 32) & 0x01ffffffu) | (2u << 30);  // type=2
        // D# group 1: data_size=4B; tensor 2048x1; tile 2048x1; stride0=2048
        i32x8 g1;
        g1[0] = (int)(2u << 16);           // workgroup_mask=0, data_size=2(4B)
        g1[1] = (int)(2048u << 16);        // tensor_dim0[15:0] in [31:16]
        g1[2] = (int)(1u << 16);           // tensor_dim0 hi=0, tensor_dim1=1
        g1[3] = (int)(2048u << 16);        // tensor_dim1 hi=0, tile_dim0=2048
        g1[4] = 1;                         // tile_dim1=1, tile_dim2=0
        g1[5] = 2048;                      // tensor_dim0_stride[31:0]
        g1[6] = 0;                         // stride0 hi, stride1 lo
        g1[7] = 0;
        i32x4 z4;
        z4[0] = 0; z4[1] = 0; z4[2] = 0; z4[3] = 0;
        i32x8 z8;
#pragma unroll
        for (int i = 0; i < 8; ++i) z8[i] = 0;
        // clang-23 / therock-10.0 lane: 6-arg form
        // (g0, g1, g2, g3, extra_group, cpol)
        __builtin_amdgcn_tensor_load_to_lds(g0, g1, z4, z4, z8, 0);
        __builtin_amdgcn_s_wait_tensorcnt(0);
    }
    __syncthreads();
#else
    for (int i = tid; i < S_LEN; i += 256)
        buf[i] = scores[(size_t)row * S_LEN + i];
    __syncthreads();
#endif

    for (int ksz = 2; ksz <= S_LEN; ksz <<= 1) {
        for (int j = ksz >> 1; j > 0; j >>= 1) {
            for (int i = tid; i < S_LEN; i += 256) {
                int l = i ^ j;
                if (l > i) {
                    bool asc = ((i & ksz) == 0);
                    float a = buf[i], c = buf[l];
                    if ((a > c) == asc) { buf[i] = c; buf[l] = a; }
                }
            }
            __syncthreads();
        }
    }
    if (tid == 0) {
        int kk = adk[row >> 11];           // row / S_LEN
        thr[row] = buf[S_LEN - kk];        // ascending: k-th largest
    }
}

// ---------------------------------------------------------------------------
// 5) rmsnorm over c = ckv[:, :512] with kv_norm_w -> cnorm (packed 512)
// ---------------------------------------------------------------------------
__global__ void rmsnorm_kernel(const float* __restrict__ ckv,
                               const float* __restrict__ w,
                               float* __restrict__ cnorm) {
    const int row = blockIdx.x, tid = threadIdx.x;
    __shared__ float red[256];
    const float* x = ckv + (size_t)row * (KVR + DR);
    float ss = 0.0f;
    for (int i = tid; i < KVR; i += 256) { float v = x[i]; ss += v * v; }
    red[tid] = ss;
    __syncthreads();
    for (int st = 128; st > 0; st >>= 1) {
        if (tid < st) red[tid] += red[tid + st];
        __syncthreads();
    }
    float r = rsqrtf(red[0] / (float)KVR + 1e-6f);
    for (int i = tid; i < KVR; i += 256)
        cnorm[(size_t)row * KVR + i] = x[i] * r * w[i];
}

// ---------------------------------------------------------------------------
// 6) RoPE in place on q_rope (qbuf[..., h*192+128 .. +191]) and k_rope
//    (ckv[..., 512..575]). Thread owns the pair (j, j+32) -> no races.
// ---------------------------------------------------------------------------
__global__ void rope_kernel(float* __restrict__ qbuf, float* __restrict__ ckv) {
    const int row = blockIdx.x;           // b*S + s
    const int pos = row & (S_LEN - 1);    // s
    const int t = threadIdx.x;            // 0..511
    const int h = t >> 5;
    const int j = t & 31;
    float inv = powf(10000.0f, -(float)(2 * j) / 64.0f);
    float ang = (float)pos * inv;
    float cs = cosf(ang), sn = sinf(ang);

    float* qp = qbuf + (size_t)row * (NH * (DN + DR)) + h * (DN + DR) + DN;
    float x1 = qp[j], x2 = qp[j + 32];
    qp[j]      = x1 * cs - x2 * sn;
    qp[j + 32] = x2 * cs + x1 * sn;
    if (t < 32) {
        float* kp = ckv + (size_t)row * (KVR + DR) + KVR;
        float y1 = kp[j], y2 = kp[j + 32];
        kp[j]      = y1 * cs - y2 * sn;
        kp[j + 32] = y2 * cs + y1 * sn;
    }
}

// ---------------------------------------------------------------------------
// 7) Flash attention: one wave per (b, h, 16-row q tile). Streams ALL k tiles
//    (mask = min(causal, sparse) as additive -1e9, matching the reference's
//    full-row softmax semantics). Online softmax; P staged through LDS to
//    become an A-fragment for the PV WMMA.
// ---------------------------------------------------------------------------
__global__ void flash_attn_kernel(const float* __restrict__ qbuf,
                                  const float* __restrict__ kvbuf,
                                  const float* __restrict__ ckv,
                                  const float* __restrict__ sidx,
                                  const float* __restrict__ thr,
                                  float* __restrict__ attn_out) {
    const int lane = threadIdx.x & 31, wave = threadIdx.x >> 5;
    const int lrow = lane & 15, lhk = lane >> 4;
    const int qt = blockIdx.x * 4 + wave;
    const int h = blockIdx.y, b = blockIdx.z;
    const int q0 = qt * 16;
    __shared__ float pshare[4][16 * 16];
    float* P = pshare[wave];
    const float scale = 0.072168783648703332f;  // 1/sqrt(DN+DR)
    const int ldq = NH * (DN + DR);   // 3072
    const int ldkv = NH * (DN + DV);  // 4096
    const int ldck = KVR + DR;        // 576

    const float* qrow = qbuf + (size_t)(b * S_LEN + q0 + lrow) * ldq + h * (DN + DR) + 2 * lhk;

    float thr_j[8];
    const float* sidx_j[8];
#pragma unroll
    for (int j = 0; j < 8; ++j) {
        int q = q0 + j + 8 * lhk;
        thr_j[j] = thr[b * S_LEN + q];
        sidx_j[j] = sidx + (size_t)(b * S_LEN + q) * S_LEN;
    }

    v8f o[8];
#pragma unroll
    for (int t = 0; t < 8; ++t) o[t] = v8f_zero();
    float m_j[8], l_j[8];
#pragma unroll
    for (int j = 0; j < 8; ++j) { m_j[j] = -3.0e38f; l_j[j] = 0.0f; }

    for (int kt = 0; kt < S_LEN / 16; ++kt) {
        const int kbase = kt * 16;
        // ---- scores tile: q_full(16x192) . k_full(16x192)^T ----
        v8f s = v8f_zero();
        const float* knrow = kvbuf + (size_t)(b * S_LEN + kbase + lrow) * ldkv + h * (DN + DV) + 2 * lhk;
        const float* krrow = ckv + (size_t)(b * S_LEN + kbase + lrow) * ldck + KVR + 2 * lhk;
#pragma unroll
        for (int d = 0; d < DN; d += 4) {
            v2f a  = *(const v2f*)(qrow + d);
            v2f bf = *(const v2f*)(knrow + d);
            s = wmma_f32(a, bf, s);
        }
#pragma unroll
        for (int d = 0; d < DR; d += 4) {
            v2f a  = *(const v2f*)(qrow + DN + d);
            v2f bf = *(const v2f*)(krrow + d);
            s = wmma_f32(a, bf, s);
        }
        // ---- scale + combined mask (min(causal, sparse) as 0/-1e9) ----
#pragma unroll
        for (int j = 0; j < 8; ++j) {
            int q = q0 + j + 8 * lhk;
            int k = kbase + lrow;
            float si = sidx_j[j][k];
            float add = ((k <= q) && (si >= thr_j[j])) ? 0.0f : NEGV;
            s[j] = s[j] * scale + add;
        }
        // ---- online softmax (row reductions across the 16-lane half-group)
        float p[8];
#pragma unroll
        for (int j = 0; j < 8; ++j) {
            float rm = s[j];
            for (int msk = 1; msk <= 8; msk <<= 1)
                rm = fmaxf(rm, __shfl_xor(rm, msk, 32));
            float mn = fmaxf(m_j[j], rm);
            float corr = expf(m_j[j] - mn);
            float pe = expf(s[j] - mn);
            float rs = pe;
            for (int msk = 1; msk <= 8; msk <<= 1)
                rs += __shfl_xor(rs, msk, 32);
            l_j[j] = l_j[j] * corr + rs;
            m_j[j] = mn;
#pragma unroll
            for (int t = 0; t < 8; ++t) o[t][j] *= corr;
            p[j] = pe;
        }
        // ---- P through LDS -> A fragment; o += P(16x16) @ V(16x128) ----
        __syncthreads();
#pragma unroll
        for (int j = 0; j < 8; ++j) P[(j + 8 * lhk) * 16 + lrow] = p[j];
        __syncthreads();
#pragma unroll
        for (int ks = 0; ks < 16; ks += 4) {
            v2f a = *(const v2f*)(P + lrow * 16 + ks + 2 * lhk);
            const float* vrow = kvbuf +
                (size_t)(b * S_LEN + kbase + ks + 2 * lhk) * ldkv +
                h * (DN + DV) + DN + lrow;
#pragma unroll
            for (int t = 0; t < 8; ++t) {
                v2f bf;
                bf.x = vrow[t * 16];
                bf.y = vrow[ldkv + t * 16];
                o[t] = wmma_f32(a, bf, o[t]);
            }
        }
    }
    // ---- finalize: attn_out[b,q, h*128 + n] = o / l ----
#pragma unroll
    for (int t = 0; t < 8; ++t) {
#pragma unroll
        for (int j = 0; j < 8; ++j) {
            int row = b * S_LEN + q0 + j + 8 * lhk;
            attn_out[(size_t)row * (NH * DV) + h * DV + t * 16 + lrow] =
                o[t][j] / l_j[j];
        }
    }
}

// ---------------------------------------------------------------------------
// Host launch. Workspace need: ~211 MB.
// ---------------------------------------------------------------------------
extern "C" void kernel_launch(void* const* d_in, const int* in_sizes, int n_in,
                              void* d_out, int out_size, void* d_ws, size_t ws_size,
                              hipStream_t stream) {
    const float* hs     = (const float*)d_in[0];
    const float* Wq_idx = (const float*)d_in[1];
    const float* bq_idx = (const float*)d_in[2];
    const float* Wk_idx = (const float*)d_in[3];
    const float* bk_idx = (const float*)d_in[4];
    const float* idx_w  = (const float*)d_in[5];
    const float* w_ctrl = (const float*)d_in[6];
    const float* b_ctrl = (const float*)d_in[7];
    const float* Wq     = (const float*)d_in[8];
    const float* W_dkv  = (const float*)d_in[9];
    const float* kv_nw  = (const float*)d_in[10];
    const float* W_ukv  = (const float*)d_in[11];
    const float* Wo     = (const float*)d_in[12];
    float* out = (float*)d_out;
    (void)in_sizes; (void)n_in; (void)out_size; (void)ws_size;

    char* base = (char*)d_ws;
    size_t off = 0;
    auto alloc = [&](size_t bytes) -> char* {
        char* r = base + off;
        off = (off + bytes + 255) & ~(size_t)255;
        return r;
    };
    const size_t MR = (size_t)B_SZ * S_LEN;  // 4096 rows
    float* qi     = (float*)alloc(MR * (HI_N * DI_N) * 4);
    float* ki     = (float*)alloc(MR * (HI_N * DI_N) * 4);
    float* sidx   = (float*)alloc(MR * S_LEN * 4);
    float* thr    = (float*)alloc(MR * 4);
    int*   adk    = (int*)  alloc(B_SZ * 4);
    float* qbuf   = (float*)alloc(MR * (NH * (DN + DR)) * 4);
    float* ckv    = (float*)alloc(MR * (KVR + DR) * 4);
    float* cnorm  = (float*)alloc(MR * KVR * 4);
    float* kvbuf  = (float*)alloc(MR * (NH * (DN + DV)) * 4);
    float* attn_o = (float*)alloc(MR * (NH * DV) * 4);

    const dim3 blk128(128);

    ctrl_kernel<<<dim3(B_SZ), dim3(256), 0, stream>>>(hs, w_ctrl, b_ctrl, adk);

    // M=4096 -> grid.x = M/32 = 128 (wave tile 32x64, block tile 32x256)
    gemm_wmma<<<dim3(128, 1), blk128, 0, stream>>>(hs, Wq_idx, bq_idx, qi, 4096, 256, 2048);
    gemm_wmma<<<dim3(128, 1), blk128, 0, stream>>>(hs, Wk_idx, bk_idx, ki, 4096, 256, 2048);

    index_scores_kernel<<<dim3(S_LEN / 16, S_LEN / 64, B_SZ), blk128, 0, stream>>>(qi, ki, idx_w, sidx);

    topk_thr_kernel<<<dim3((unsigned)MR), dim3(256), 0, stream>>>(sidx, adk, thr);

    gemm_wmma<<<dim3(128, 12), blk128, 0, stream>>>(hs, Wq, (const float*)nullptr, qbuf, 4096, 3072, 2048);
    gemm_wmma<<<dim3(128, 3),  blk128, 0, stream>>>(hs, W_dkv, (const float*)nullptr, ckv, 4096, 576, 2048);

    rmsnorm_kernel<<<dim3((unsigned)MR), dim3(256), 0, stream>>>(ckv, kv_nw, cnorm);

    gemm_wmma<<<dim3(128, 16), blk128, 0, stream>>>(cnorm, W_ukv, (const float*)nullptr, kvbuf, 4096, 4096, 512);

    rope_kernel<<<dim3((unsigned)MR), dim3(512), 0, stream>>>(qbuf, ckv);

    flash_attn_kernel<<<dim3(S_LEN / 64, NH, B_SZ), blk128, 0, stream>>>(qbuf, kvbuf, ckv, sidx, thr, attn_o);

    gemm_wmma<<<dim3(128, 8), blk128, 0, stream>>>(attn_o, Wo, (const float*)nullptr, out, 4096, 2048, 2048);
}